// RetinalContrastGraphBuilder_57672820851269
// MI455X (gfx1250) — compile-verified
//
#include <hip/hip_runtime.h>
#include <hip/hip_bf16.h>
#include <math.h>

// ---------------------------------------------------------------------------
// RetinalContrastGraphBuilder for MI455X (gfx1250, wave32, WMMA)
//
// Pipeline:  crop64+gray+3xDoG conv -> per-batch top-196 -> node features ->
//            196x196 pairwise distance via V_WMMA_F32_16X16X4_F32 Gram matrix
//            -> 6-NN selection -> edge features.
// ---------------------------------------------------------------------------

#define NBATCH       64
#define IMG_H        512
#define IMG_W        512
#define CROPN        64
#define MAX_NODES    196
#define PAD_NODES    208          // 13 tiles of 16
#define NTILES       13
#define KNEIGH       6
#define EVENT_THRESH 0.1f
#define DIST_THRESH  (4.05f / 4.2f)
#define BIGF         1000000000.0f
#define FLAT_N       (3 * CROPN * CROPN)   // 12288

// output layout (floats, concatenated in reference return order)
#define OFF_NODES   0            // 64*196*5             = 62720
#define OFF_EIDX    62720        // 2 * 64*196*6         = 150528
#define OFF_EFEAT   213248       // 64*196*6*3           = 225792
#define OFF_BATCH   439040       // 64*196               = 12544
#define OFF_NVALID  451584       // 64*196               = 12544
#define OFF_EV      464128       // 64*196*6             = 75264
#define NEDGE_TOT   75264

typedef __attribute__((ext_vector_type(2))) float v2f;
typedef __attribute__((ext_vector_type(8))) float v8f;

// ---------------------------------------------------------------------------
// Kernel A: bilinear crop -> gray -> 3 DoG convolutions, per-batch block.
// ---------------------------------------------------------------------------
__global__ void __launch_bounds__(256)
crop_conv_kernel(const float* __restrict__ image,
                 const float* __restrict__ w1,
                 const float* __restrict__ w2,
                 const float* __restrict__ w3,
                 float* __restrict__ resp) {
    __shared__ float gray[CROPN * (CROPN + 1)];   // +1 pad vs 64-bank LDS
    __shared__ float w1s[9], w2s[25], w3s[81];
    __shared__ int   si[CROPN];
    __shared__ float sf[CROPN];

    const int b   = blockIdx.x;
    const int tid = threadIdx.x;

    // CDNA5 global_prefetch_b8 of the only image rows we touch (247..264, 3ch)
    if (tid < 54) {
        int c = tid / 18, y = 247 + (tid % 18);
        __builtin_prefetch(image + (((size_t)b * 3 + c) * IMG_H + y) * IMG_W + 247, 0, 3);
    }

    if (tid < 9)  w1s[tid] = w1[tid];
    if (tid < 25) w2s[tid] = w2[tid];
    if (tid < 81) w3s[tid] = w3[tid];
    if (tid < CROPN) {
        float base = ((float)tid + 0.5f) / 32.0f - 1.0f;   // (p+0.5)/(CROP/2)-1
        float g    = base * (16.0f / 512.0f);
        float p    = (g + 1.0f) * 256.0f - 0.5f;
        float fl   = floorf(p);
        si[tid] = (int)fl;
        sf[tid] = p - fl;
    }
    __syncthreads();

    // bilinear gather, channel mean
    for (int idx = tid; idx < CROPN * CROPN; idx += 256) {
        int iy = idx >> 6, ix = idx & 63;
        int   x0 = si[ix];  float fx = sf[ix];
        int   y0 = si[iy];  float fy = sf[iy];
        int   x1 = x0 + 1,  y1 = y0 + 1;
        float vx0 = (x0 >= 0 && x0 < IMG_W) ? 1.f : 0.f;
        float vx1 = (x1 >= 0 && x1 < IMG_W) ? 1.f : 0.f;
        float vy0 = (y0 >= 0 && y0 < IMG_H) ? 1.f : 0.f;
        float vy1 = (y1 >= 0 && y1 < IMG_H) ? 1.f : 0.f;
        int cx0 = min(max(x0, 0), IMG_W - 1), cx1 = min(max(x1, 0), IMG_W - 1);
        int cy0 = min(max(y0, 0), IMG_H - 1), cy1 = min(max(y1, 0), IMG_H - 1);
        float wx0 = (1.f - fx) * vx0, wx1 = fx * vx1;
        float wy0 = (1.f - fy) * vy0, wy1 = fy * vy1;
        float g = 0.f;
        #pragma unroll
        for (int c = 0; c < 3; ++c) {
            const float* p = image + (((size_t)b * 3 + c) * IMG_H) * IMG_W;
            float a00 = p[(size_t)cy0 * IMG_W + cx0];
            float a10 = p[(size_t)cy0 * IMG_W + cx1];
            float a01 = p[(size_t)cy1 * IMG_W + cx0];
            float a11 = p[(size_t)cy1 * IMG_W + cx1];
            g += (a00 * wx0 + a10 * wx1) * wy0 + (a01 * wx0 + a11 * wx1) * wy1;
        }
        gray[iy * (CROPN + 1) + ix] = g * (1.f / 3.f);
    }
    __syncthreads();

    // 3 'SAME' (zero-pad) DoG convolutions out of LDS
    float* rb = resp + (size_t)b * FLAT_N;
    for (int idx = tid; idx < CROPN * CROPN; idx += 256) {
        int iy = idx >> 6, ix = idx & 63;
        float r1 = 0.f, r2 = 0.f, r3 = 0.f;
        #pragma unroll
        for (int dy = -4; dy <= 4; ++dy) {
            int  y   = iy + dy;
            bool yok = (y >= 0) && (y < CROPN);
            #pragma unroll
            for (int dx = -4; dx <= 4; ++dx) {
                int   x = ix + dx;
                float t = (yok && x >= 0 && x < CROPN) ? gray[y * (CROPN + 1) + x] : 0.f;
                r3 += t * w3s[(dy + 4) * 9 + (dx + 4)];
                if (dy >= -2 && dy <= 2 && dx >= -2 && dx <= 2)
                    r2 += t * w2s[(dy + 2) * 5 + (dx + 2)];
                if (dy >= -1 && dy <= 1 && dx >= -1 && dx <= 1)
                    r1 += t * w1s[(dy + 1) * 3 + (dx + 1)];
            }
        }
        rb[idx]        = r1;   // channel 0
        rb[4096 + idx] = r2;   // channel 1
        rb[8192 + idx] = r3;   // channel 2
    }
}

// ---------------------------------------------------------------------------
// Kernel B: per-batch top-196 of score = (|r| > thresh ? |r| : -1).
// LDS-resident scores, 196 argmax passes with a wave32-friendly tree reduce.
// Tie-break: lower index first (matches jax.lax.top_k).
// ---------------------------------------------------------------------------
__global__ void __launch_bounds__(256)
topk_kernel(const float* __restrict__ resp,
            float* __restrict__ tv, int* __restrict__ ti) {
    __shared__ float sc[FLAT_N];
    __shared__ float rv[256];
    __shared__ int   ri[256];

    const int b   = blockIdx.x;
    const int tid = threadIdx.x;
    const float* r = resp + (size_t)b * FLAT_N;

    for (int i = tid; i < FLAT_N; i += 256) {
        float m = fabsf(r[i]);
        sc[i] = (m > EVENT_THRESH) ? m : -1.0f;
    }
    __syncthreads();

    for (int k = 0; k < MAX_NODES; ++k) {
        float bv = -5.0f; int bi = FLAT_N;
        for (int i = tid; i < FLAT_N; i += 256) {
            float v = sc[i];
            if (v > bv || (v == bv && i < bi)) { bv = v; bi = i; }
        }
        rv[tid] = bv; ri[tid] = bi;
        __syncthreads();
        for (int s = 128; s > 0; s >>= 1) {
            if (tid < s) {
                float ov = rv[tid + s]; int oi = ri[tid + s];
                if (ov > rv[tid] || (ov == rv[tid] && oi < ri[tid])) {
                    rv[tid] = ov; ri[tid] = oi;
                }
            }
            __syncthreads();
        }
        if (tid == 0) {
            tv[b * MAX_NODES + k] = rv[0];
            ti[b * MAX_NODES + k] = ri[0];
            sc[ri[0]] = -4.0f;   // below any live score; never re-selected
        }
        __syncthreads();
    }
}

// ---------------------------------------------------------------------------
// Kernel C: node features, WMMA Gram matrix -> pairwise distances, 6-NN,
// edge features. One 256-thread (8 wave32) block per batch.
// Gram tile: D = A(16x4 f32) x B(4x16 f32) via v_wmma_f32_16x16x4_f32;
// coordinates are 2-D so K=2,3 lanes are zero-padded.
// ---------------------------------------------------------------------------
__global__ void __launch_bounds__(256)
graph_kernel(const float* __restrict__ resp,
             const float* __restrict__ tv, const int* __restrict__ ti,
             float* __restrict__ out) {
    __shared__ float cx[PAD_NODES], cy[PAD_NODES], cnorm[PAD_NODES];
    __shared__ int   cvalid[PAD_NODES];
    __shared__ float dmat[16 * PAD_NODES];   // one 16-row distance slab

    const int b   = blockIdx.x;
    const int tid = threadIdx.x;

    float* out_nodes  = out + OFF_NODES;
    float* out_eidx   = out + OFF_EIDX;     // row 0: src, row 1: dst (+NEDGE_TOT)
    float* out_efeat  = out + OFF_EFEAT;
    float* out_batch  = out + OFF_BATCH;
    float* out_nvalid = out + OFF_NVALID;
    float* out_ev     = out + OFF_EV;

    // ---- Phase 1: node features + masked coords into LDS -------------------
    if (tid < PAD_NODES) {
        float x = 0.f, y = 0.f; int valid = 0;
        if (tid < MAX_NODES) {
            int   gi  = b * MAX_NODES + tid;
            float val = tv[gi];
            int   idx = ti[gi];
            valid = (val > EVENT_THRESH) ? 1 : 0;
            int c   = idx / 4096;
            int rem = idx - c * 4096;
            int yi  = rem >> 6;
            int xi  = rem & 63;
            float xc  = (float)xi * (2.0f / 63.0f) - 1.0f;
            float yc  = (float)yi * (2.0f / 63.0f) - 1.0f;
            float ecc = sqrtf(xc * xc + yc * yc);
            float rvv = resp[(size_t)b * FLAT_N + idx];
            float pol = (rvv > 0.f) ? 1.f : ((rvv < 0.f) ? -1.f : 0.f);
            float m   = (float)valid;
            size_t o5 = (size_t)gi * 5;
            out_nodes[o5 + 0] = xc * m;
            out_nodes[o5 + 1] = yc * m;
            out_nodes[o5 + 2] = pol * m;
            out_nodes[o5 + 3] = (float)c * m;
            out_nodes[o5 + 4] = ecc * m;
            out_batch[gi]  = (float)b;
            out_nvalid[gi] = m;
            x = xc * m; y = yc * m;
        }
        cx[tid] = x; cy[tid] = y;
        cvalid[tid] = valid;
        cnorm[tid]  = x * x + y * y;
    }
    __syncthreads();

    const int wave = tid >> 5;    // 8 wave32's
    const int lane = tid & 31;
    const int half = lane >> 4;   // 0: lanes 0-15 (K=0,1), 1: lanes 16-31 (K=2,3 pad)
    const int l16  = lane & 15;

    // ---- Phase 2: per 16-row tile: WMMA distances, then 6-NN select --------
    for (int mt = 0; mt < NTILES; ++mt) {
        for (int nt = wave; nt < NTILES; nt += 8) {
            // A 16x4: row M=l16, VGPR0/1 = K0/K1 in lanes 0-15; K2/K3 zeros.
            v2f a, bb;
            a.x  = (half == 0) ? cx[mt * 16 + l16] : 0.f;
            a.y  = (half == 0) ? cy[mt * 16 + l16] : 0.f;
            // B 4x16: col N=l16, K rows mirrored to A's lane-half split.
            bb.x = (half == 0) ? cx[nt * 16 + l16] : 0.f;
            bb.y = (half == 0) ? cy[nt * 16 + l16] : 0.f;
            v8f acc = {};
            acc = __builtin_amdgcn_wmma_f32_16x16x4_f32(
                      /*neg_a=*/false, a, /*neg_b=*/false, bb,
                      /*c_mod=*/(short)0, acc, /*reuse_a=*/false, /*reuse_b=*/false);
            #pragma unroll
            for (int r = 0; r < 8; ++r) {
                int mloc = r + 8 * half;            // C/D layout: VGPR r, halves
                int gi = mt * 16 + mloc;
                int gj = nt * 16 + l16;
                float d2   = cnorm[gi] + cnorm[gj] - 2.f * acc[r];
                float dist = (d2 > 0.f) ? sqrtf(d2) : 0.f;
                int ok = cvalid[gi] & cvalid[gj] & (int)(gi != gj);
                dmat[mloc * PAD_NODES + gj] = ok ? dist : BIGF;
            }
        }
        __syncthreads();

        // 6 smallest per row (ties -> lower index, like top_k of negation)
        if (tid < 16) {
            int i = mt * 16 + tid;
            if (i < MAX_NODES) {
                float bd[KNEIGH]; int bj[KNEIGH];
                #pragma unroll
                for (int k = 0; k < KNEIGH; ++k) { bd[k] = BIGF; bj[k] = 0; }
                for (int j = 0; j < PAD_NODES; ++j) {
                    float d = dmat[tid * PAD_NODES + j];
                    if (d < bd[KNEIGH - 1]) {
                        int k = KNEIGH - 1;
                        while (k > 0 && d < bd[k - 1]) {
                            bd[k] = bd[k - 1]; bj[k] = bj[k - 1]; --k;
                        }
                        bd[k] = d; bj[k] = j;
                    }
                }
                int   srcv = cvalid[i];
                float sx = cx[i], sy = cy[i];
                #pragma unroll
                for (int k = 0; k < KNEIGH; ++k) {
                    int  nbr = bj[k];
                    int  ev  = (bd[k] < DIST_THRESH && srcv) ? 1 : 0;
                    int  e   = (b * MAX_NODES + i) * KNEIGH + k;
                    float fx = (float)ev;
                    float dx = cx[nbr] - sx;
                    float dy = cy[nbr] - sy;
                    float s2 = dx * dx + dy * dy;
                    float cd = (s2 > 0.f) ? sqrtf(s2) : 0.f;
                    out_eidx[e]             = ev ? (float)(i + b * MAX_NODES)   : 0.f;
                    out_eidx[NEDGE_TOT + e] = ev ? (float)(nbr + b * MAX_NODES) : 0.f;
                    out_efeat[(size_t)e * 3 + 0] = dx * fx;
                    out_efeat[(size_t)e * 3 + 1] = dy * fx;
                    out_efeat[(size_t)e * 3 + 2] = cd * fx;
                    out_ev[e] = fx;
                }
            }
        }
        __syncthreads();   // dmat slab reused next row tile
    }
}

// ---------------------------------------------------------------------------
extern "C" void kernel_launch(void* const* d_in, const int* in_sizes, int n_in,
                              void* d_out, int out_size, void* d_ws, size_t ws_size,
                              hipStream_t stream) {
    const float* image = (const float*)d_in[0];
    const float* w1    = (const float*)d_in[1];
    const float* w2    = (const float*)d_in[2];
    const float* w3    = (const float*)d_in[3];
    float* out = (float*)d_out;

    // workspace layout
    float* resp = (float*)d_ws;                        // 64*12288 f32
    float* tv   = resp + (size_t)NBATCH * FLAT_N;      // 64*196   f32
    int*   ti   = (int*)(tv + NBATCH * MAX_NODES);     // 64*196   i32

    crop_conv_kernel<<<NBATCH, 256, 0, stream>>>(image, w1, w2, w3, resp);
    topk_kernel<<<NBATCH, 256, 0, stream>>>(resp, tv, ti);
    graph_kernel<<<NBATCH, 256, 0, stream>>>(resp, tv, ti, out);
}